// AdaLoRAWithBase_84121229459560
// MI455X (gfx1250) — compile-verified
//
#include <hip/hip_runtime.h>
#include <math.h>

typedef __attribute__((ext_vector_type(2))) float v2f;
typedef __attribute__((ext_vector_type(8))) float v8f;

#define LN_EPS 1e-5f
constexpr int Bn = 128, Tn = 32, Dn = 1024, ADAn = 1024, INTERn = 1024, RANKn = 8;
constexpr int BK = 16;        // K chunk staged in LDS
constexpr int LPAD = 20;      // padded LDS row stride (floats): 80B, 16B-aligned, conflict-free frags

// ---------------------------------------------------------------------------
// LayerNorm over last dim (ADA=1024), one block per row
// ---------------------------------------------------------------------------
__global__ void ln_kernel(const float* __restrict__ ada, const float* __restrict__ g,
                          const float* __restrict__ bb, float* __restrict__ out) {
  const int row = blockIdx.x;
  const float* in = ada + (size_t)row * ADAn;
  float s = 0.f, s2 = 0.f;
  for (int i = threadIdx.x; i < ADAn; i += blockDim.x) {
    float v = in[i]; s += v; s2 += v * v;
  }
  __shared__ float red[64];
  for (int off = 16; off > 0; off >>= 1) {
    s  += __shfl_down(s,  off, 32);
    s2 += __shfl_down(s2, off, 32);
  }
  const int wave = threadIdx.x >> 5, lane = threadIdx.x & 31;
  const int nw = blockDim.x >> 5;
  if (lane == 0) { red[wave] = s; red[32 + wave] = s2; }
  __syncthreads();
  if (wave == 0) {
    float a  = (lane < nw) ? red[lane] : 0.f;
    float a2 = (lane < nw) ? red[32 + lane] : 0.f;
    for (int off = 16; off > 0; off >>= 1) {
      a  += __shfl_down(a,  off, 32);
      a2 += __shfl_down(a2, off, 32);
    }
    if (lane == 0) { red[0] = a; red[1] = a2; }
  }
  __syncthreads();
  const float mean = red[0] * (1.f / ADAn);
  const float var  = red[1] * (1.f / ADAn) - mean * mean;
  const float rstd = rsqrtf(var + LN_EPS);
  for (int i = threadIdx.x; i < ADAn; i += blockDim.x)
    out[(size_t)row * ADAn + i] = (in[i] - mean) * rstd * g[i] + bb[i];
}

// ---------------------------------------------------------------------------
// Shared WMMA fp32 GEMM core: one 64x64 block tile of A(MxK,row) @ W(KxN,row).
// 4 waves; wave w owns rows [w*16, w*16+16) x all 64 cols -> 4 x v8f acc.
// A tile:  global -> LDS via global_load_async_to_lds_b128 (ASYNCcnt DMA,
//          no VGPR round-trip, no s_wait_loadcnt stall).
// B tile:  VGPR-staged + transposed into LDS; global loads for chunk k+1 are
//          issued before the WMMA phase of chunk k (software pipeline).
// A fragment (16x4): lane l -> m = l&15, k = kc + 2*(l>>4) + {0,1}
// B fragment (4x16): lane l -> n = l&15, k = kc + 2*(l>>4) + {0,1} (LDS = B^T)
// ---------------------------------------------------------------------------
__device__ __forceinline__ void gemm_core_64x64(const float* __restrict__ A,
                                                const float* __restrict__ W,
                                                int K, int N, int rowBase, int colBase,
                                                v8f acc[4]) {
  __shared__ float lA [64][LPAD];
  __shared__ float lBt[64][LPAD];
  const int tid  = threadIdx.x, lane = tid & 31;
  const int mBase = (tid >> 5) * 16;
  const int mrow = lane & 15, khalf = lane >> 4;

  // per-thread staging coordinates (2 x 16B chunks each for A and B)
  const int l0  = tid * 2;
  const int ar0 = l0 >> 2,       ac0 = (l0 & 3) * 4;         // A: 4 chunks/row
  const int ar1 = (l0 + 1) >> 2, ac1 = ((l0 + 1) & 3) * 4;
  const int kk0 = l0 >> 4,       n40 = (l0 & 15) * 4;        // B: 16 chunks/k-row
  const int kk1 = (l0 + 1) >> 4, n41 = ((l0 + 1) & 15) * 4;

  // LDS byte addresses for the async A chunks (generic addr low 32 bits = LDS offset)
  const unsigned ldsA0 = (unsigned)(size_t)&lA[ar0][ac0];
  const unsigned ldsA1 = (unsigned)(size_t)&lA[ar1][ac1];

  // prologue: issue B global loads for chunk 0
  float4 rb0 = *(const float4*)(W + (size_t)kk0 * N + colBase + n40);
  float4 rb1 = *(const float4*)(W + (size_t)kk1 * N + colBase + n41);

  for (int k0 = 0; k0 < K; k0 += BK) {
    __syncthreads();                                   // all waves done reading LDS
    // ---- A tile: async DMA straight into (padded) LDS -------------------
    {
      const float* ga0 = A + (size_t)(rowBase + ar0) * K + k0 + ac0;
      const float* ga1 = A + (size_t)(rowBase + ar1) * K + k0 + ac1;
      asm volatile("global_load_async_to_lds_b128 %0, %1, off"
                   :: "v"(ldsA0), "v"(ga0) : "memory");
      asm volatile("global_load_async_to_lds_b128 %0, %1, off"
                   :: "v"(ldsA1), "v"(ga1) : "memory");
    }
    // ---- B tile: transpose staged regs into LDS (loads issued last iter) -
    lBt[n40 + 0][kk0] = rb0.x; lBt[n40 + 1][kk0] = rb0.y;
    lBt[n40 + 2][kk0] = rb0.z; lBt[n40 + 3][kk0] = rb0.w;
    lBt[n41 + 0][kk1] = rb1.x; lBt[n41 + 1][kk1] = rb1.y;
    lBt[n41 + 2][kk1] = rb1.z; lBt[n41 + 3][kk1] = rb1.w;
    asm volatile("s_wait_asynccnt 0x0" ::: "memory");  // A DMA landed
    __syncthreads();
    // ---- issue B global loads for NEXT chunk (consumed next iteration) ---
    if (k0 + BK < K) {
      rb0 = *(const float4*)(W + (size_t)(k0 + BK + kk0) * N + colBase + n40);
      rb1 = *(const float4*)(W + (size_t)(k0 + BK + kk1) * N + colBase + n41);
      __builtin_prefetch(A + (size_t)(rowBase + ar0) * K + k0 + BK + ac0, 0, 1);
    }
    // ---- compute: 16 x v_wmma_f32_16x16x4_f32 per wave -------------------
    #pragma unroll
    for (int ks = 0; ks < 4; ++ks) {
      const int kc = ks * 4 + 2 * khalf;
      const v2f a = *(const v2f*)&lA[mBase + mrow][kc];
      #pragma unroll
      for (int nt = 0; nt < 4; ++nt) {
        const v2f b = *(const v2f*)&lBt[nt * 16 + mrow][kc];
        acc[nt] = __builtin_amdgcn_wmma_f32_16x16x4_f32(
            false, a, false, b, (short)0, acc[nt], false, false);
      }
    }
  }
}

// ---------------------------------------------------------------------------
// Hypernetwork GEMM: out = act(A @ W + bias); act: 1 = exact GELU, 0 = identity
// ---------------------------------------------------------------------------
__global__ __launch_bounds__(128) void hyper_gemm(const float* __restrict__ A,
                                                  const float* __restrict__ W,
                                                  const float* __restrict__ bias,
                                                  float* __restrict__ out,
                                                  int K, int N, int gelu) {
  const int rowBase = blockIdx.y * 64, colBase = blockIdx.x * 64;
  v8f acc[4] = {};
  gemm_core_64x64(A, W, K, N, rowBase, colBase, acc);

  const int lane = threadIdx.x & 31, wave = threadIdx.x >> 5;
  #pragma unroll
  for (int nt = 0; nt < 4; ++nt) {
    #pragma unroll
    for (int v = 0; v < 8; ++v) {
      const int row = rowBase + wave * 16 + v + 8 * (lane >> 4);
      const int col = colBase + nt * 16 + (lane & 15);
      float r = acc[nt][v] + bias[col];
      if (gelu) r = 0.5f * r * (1.f + erff(r * 0.70710678118654752f));
      out[(size_t)row * N + col] = r;
    }
  }
}

// ---------------------------------------------------------------------------
// p[b,t,r] = sum_c x[b,t,c] * x_a[b,c,r],  x_a[b,c,r] = xw[b, c*8+r]
// ---------------------------------------------------------------------------
__global__ void p_kernel(const float* __restrict__ x, const float* __restrict__ xw,
                         float* __restrict__ p) {
  const int b = blockIdx.x;
  const int t = threadIdx.x >> 3, r = threadIdx.x & 7;
  const float* xr = x + ((size_t)b * Tn + t) * Dn;
  const float* aw = xw + (size_t)b * (2 * Dn * RANKn) + r;
  float s = 0.f;
  for (int c = 0; c < Dn; ++c) s += xr[c] * aw[(size_t)c * RANKn];
  p[((size_t)b * Tn + t) * RANKn + r] = s;
}

// ---------------------------------------------------------------------------
// Final fused GEMM: out = x + x2d @ base + p @ x_b^T
// x2d is (B*T, D); rows of a wave tile stay inside one sample (16 | 32).
// ---------------------------------------------------------------------------
__global__ __launch_bounds__(128) void final_gemm(const float* __restrict__ x2d,
                                                  const float* __restrict__ base,
                                                  const float* __restrict__ xw,
                                                  const float* __restrict__ p,
                                                  float* __restrict__ out) {
  const int rowBase = blockIdx.y * 64, colBase = blockIdx.x * 64;
  v8f acc[4] = {};
  gemm_core_64x64(x2d, base, Dn, Dn, rowBase, colBase, acc);

  const int lane = threadIdx.x & 31, wave = threadIdx.x >> 5;
  #pragma unroll
  for (int nt = 0; nt < 4; ++nt) {
    #pragma unroll
    for (int v = 0; v < 8; ++v) {
      const int row = rowBase + wave * 16 + v + 8 * (lane >> 4);
      const int col = colBase + nt * 16 + (lane & 15);
      const int b = row >> 5, t = row & 31;
      const float* pr = p + ((size_t)b * Tn + t) * RANKn;
      const float* xb = xw + (size_t)b * (2 * Dn * RANKn) + (size_t)Dn * RANKn
                           + (size_t)col * RANKn;                  // x_b[b, col, :]
      float lr = 0.f;
      #pragma unroll
      for (int r = 0; r < RANKn; ++r) lr += pr[r] * xb[r];
      out[(size_t)row * Dn + col] = x2d[(size_t)row * Dn + col] + acc[nt][v] + lr;
    }
  }
}

// ---------------------------------------------------------------------------
extern "C" void kernel_launch(void* const* d_in, const int* in_sizes, int n_in,
                              void* d_out, int out_size, void* d_ws, size_t ws_size,
                              hipStream_t stream) {
  (void)in_sizes; (void)n_in; (void)out_size; (void)ws_size;
  const float* x    = (const float*)d_in[0];   // (B,T,D)
  const float* ada  = (const float*)d_in[1];   // (B,ADA)
  const float* base = (const float*)d_in[2];   // (D,D)
  const float* w1   = (const float*)d_in[3];   // (ADA,INTER)
  const float* b1   = (const float*)d_in[4];   // (INTER)
  const float* w2   = (const float*)d_in[5];   // (INTER, 2*D*RANK)
  const float* b2   = (const float*)d_in[6];   // (2*D*RANK)
  const float* ln_g = (const float*)d_in[7];   // (ADA)
  const float* ln_b = (const float*)d_in[8];   // (ADA)
  float* out = (float*)d_out;

  float* ws    = (float*)d_ws;
  float* ae_ln = ws;                                   // 128*1024
  float* h     = ae_ln + (size_t)Bn * ADAn;            // 128*1024
  float* xw    = h + (size_t)Bn * INTERn;              // 128*16384
  float* p     = xw + (size_t)Bn * 2 * Dn * RANKn;     // 128*32*8

  // 1) LayerNorm
  ln_kernel<<<Bn, 256, 0, stream>>>(ada, ln_g, ln_b, ae_ln);
  // 2) h = GELU(ae_ln @ w1 + b1)        (128x1024x1024)
  hyper_gemm<<<dim3(INTERn / 64, Bn / 64), 128, 0, stream>>>(ae_ln, w1, b1, h, ADAn, INTERn, 1);
  // 3) xw = h @ w2 + b2                 (128x1024x16384)
  hyper_gemm<<<dim3(2 * Dn * RANKn / 64, Bn / 64), 128, 0, stream>>>(h, w2, b2, xw, INTERn, 2 * Dn * RANKn, 0);
  // 4) p = per-sample x @ x_a           (tiny, rank 8)
  p_kernel<<<Bn, 256, 0, stream>>>(x, xw, p);
  // 5) out = x + x2d @ base + p @ x_b^T (4096x1024x1024 + rank-8 epilogue)
  final_gemm<<<dim3(Dn / 64, Bn * Tn / 64), 128, 0, stream>>>(x, base, xw, p, out);
}